// GCN_12438225289268
// MI455X (gfx1250) — compile-verified
//
#include <hip/hip_runtime.h>

typedef float v2f __attribute__((ext_vector_type(2)));
typedef float v8f __attribute__((ext_vector_type(8)));

// ---------------------------------------------------------------------------
// Init: deg[i] = 1.0 (self loop), zero agg (N*128), zero out (N*64)
// ---------------------------------------------------------------------------
__global__ void gcn_init(float* __restrict__ deg, float* __restrict__ agg,
                         float* __restrict__ out, int n, long long aggElems,
                         long long outElems) {
  long long idx = (long long)blockIdx.x * blockDim.x + threadIdx.x;
  if (idx < aggElems) agg[idx] = 0.0f;
  if (idx < outElems) out[idx] = 0.0f;
  if (idx < n) deg[idx] = 1.0f;  // self-loop contribution
}

// ---------------------------------------------------------------------------
// Degree: one thread per edge, hardware f32 atomic into L2
// ---------------------------------------------------------------------------
__global__ void gcn_degree(const int* __restrict__ dst, float* __restrict__ deg,
                           long long E) {
  long long e = (long long)blockIdx.x * blockDim.x + threadIdx.x;
  if (e >= E) return;
  unsafeAtomicAdd(&deg[dst[e]], 1.0f);
}

// ---------------------------------------------------------------------------
// dinv = rsqrt(deg) in place (deg >= 1 always due to self loops)
// ---------------------------------------------------------------------------
__global__ void gcn_rsqrt(float* __restrict__ deg, int n) {
  int i = blockIdx.x * blockDim.x + threadIdx.x;
  if (i >= n) return;
  deg[i] = __frsqrt_rn(deg[i]);
}

// ---------------------------------------------------------------------------
// GEMM D[M,Np] = A[M,K] * B[K,Np] with V_WMMA_F32_16X16X4_F32 (full fp32).
// One wave per 16x16 tile; 4 waves/block share the same 16 rows of A (L1 reuse).
// Layouts per CDNA5 ISA 7.12.2:
//   A 16x4 : lane%16 = M row, (lane/16)*2 = K offset -> {a.x, a.y} = K, K+1
//   B 4x16 : lane%16 = N col, (lane/16)*2 = K offset -> {b.x, b.y} = K, K+1
//   C/D    : VGPR r, lanes 0-15 -> M=r, lanes 16-31 -> M=r+8, N = lane%16
// ---------------------------------------------------------------------------
__global__ void gcn_gemm_wmma(const float* __restrict__ A,
                              const float* __restrict__ B,
                              float* __restrict__ D, int M, int K, int Np) {
  const int lane = threadIdx.x & 31;
  const int wave = threadIdx.x >> 5;
  const int mTile = blockIdx.x;
  const int nTile = blockIdx.y * 4 + wave;
  if (nTile * 16 >= Np) return;  // uniform per wave: EXEC stays all-ones

  const int m = mTile * 16 + (lane & 15);
  const int n = nTile * 16 + (lane & 15);
  const int koff = (lane >> 4) << 1;

  const float* __restrict__ arow = A + (long long)m * K;
  v8f c = {};
  for (int k0 = 0; k0 < K; k0 += 4) {
    const int k = k0 + koff;
    v2f a, b;
    a.x = arow[k];
    a.y = arow[k + 1];
    b.x = B[(long long)k * Np + n];
    b.y = B[(long long)(k + 1) * Np + n];
    c = __builtin_amdgcn_wmma_f32_16x16x4_f32(
        /*neg_a=*/false, a, /*neg_b=*/false, b,
        /*c_mod=*/(short)0, c, /*reuse_a=*/false, /*reuse_b=*/false);
  }

  const int rowBase = mTile * 16 + ((lane >> 4) << 3);  // +8 for lanes 16-31
#pragma unroll
  for (int r = 0; r < 8; ++r) {
    D[(long long)(rowBase + r) * Np + n] = c[r];
  }
}

// ---------------------------------------------------------------------------
// Edge scatter: one wave per edge. Lane loads F/32 contiguous floats of h[src]
// (float4 => 512B coalesced per wave) and atomically adds norm-scaled values
// into agg[dst]. h and agg are L2-resident (51MB each vs 192MB L2).
// ---------------------------------------------------------------------------
__global__ void gcn_scatter(const float* __restrict__ h,
                            const int* __restrict__ src,
                            const int* __restrict__ dst,
                            const float* __restrict__ dinv,
                            float* __restrict__ agg, long long E, int F) {
  const int wavesPerBlock = blockDim.x >> 5;
  const long long e = (long long)blockIdx.x * wavesPerBlock + (threadIdx.x >> 5);
  if (e >= E) return;
  const int lane = threadIdx.x & 31;
  const int s = src[e];
  const int d = dst[e];
  const float nrm = dinv[s] * dinv[d];
  const int perLane = F >> 5;  // 4 (F=128) or 2 (F=64)
  const float* __restrict__ hs = h + (long long)s * F + lane * perLane;
  float* __restrict__ od = agg + (long long)d * F + lane * perLane;
  if (perLane == 4) {
    const float4 v = *(const float4*)hs;
    unsafeAtomicAdd(od + 0, v.x * nrm);
    unsafeAtomicAdd(od + 1, v.y * nrm);
    unsafeAtomicAdd(od + 2, v.z * nrm);
    unsafeAtomicAdd(od + 3, v.w * nrm);
  } else {
    const float2 v = *(const float2*)hs;
    unsafeAtomicAdd(od + 0, v.x * nrm);
    unsafeAtomicAdd(od + 1, v.y * nrm);
  }
}

// ---------------------------------------------------------------------------
// Layer-1 finalize: z = relu(agg + h*dinv^2 (self loop) + b0), written in place
// ---------------------------------------------------------------------------
__global__ void gcn_finalize_relu(float* __restrict__ agg,
                                  const float* __restrict__ h,
                                  const float* __restrict__ dinv,
                                  const float* __restrict__ bias, int n, int F) {
  long long idx = (long long)blockIdx.x * blockDim.x + threadIdx.x;
  if (idx >= (long long)n * F) return;
  const int i = (int)(idx >> 7);   // F == 128
  const int k = (int)(idx & 127);
  const float di = dinv[i];
  const float z = agg[idx] + h[idx] * di * di + bias[k];
  agg[idx] = fmaxf(z, 0.0f);
}

// ---------------------------------------------------------------------------
// Layer-2 finalize: out += h2*dinv^2 (self loop) + b1
// ---------------------------------------------------------------------------
__global__ void gcn_finalize_out(float* __restrict__ out,
                                 const float* __restrict__ h2,
                                 const float* __restrict__ dinv,
                                 const float* __restrict__ bias, int n, int F) {
  long long idx = (long long)blockIdx.x * blockDim.x + threadIdx.x;
  if (idx >= (long long)n * F) return;
  const int i = (int)(idx >> 6);   // F == 64
  const int k = (int)(idx & 63);
  const float di = dinv[i];
  out[idx] += h2[idx] * di * di + bias[k];
}

// ---------------------------------------------------------------------------
extern "C" void kernel_launch(void* const* d_in, const int* in_sizes, int n_in,
                              void* d_out, int out_size, void* d_ws, size_t ws_size,
                              hipStream_t stream) {
  const float* x  = (const float*)d_in[0];
  const int* ei   = (const int*)d_in[1];
  const float* W0 = (const float*)d_in[2];
  const float* b0 = (const float*)d_in[3];
  const float* W1 = (const float*)d_in[4];
  const float* b1 = (const float*)d_in[5];

  const int dHid = in_sizes[3];             // 128
  const int dOut = in_sizes[5];             // 64
  const int dIn  = in_sizes[2] / dHid;      // 128
  const int N    = in_sizes[0] / dIn;       // 100000
  const long long E = in_sizes[1] / 2;      // 1.6M
  const int* src = ei;
  const int* dst = ei + E;

  // Workspace layout (floats): dinv[N] | h[N*dHid] | agg[N*dHid]
  float* dinv = (float*)d_ws;
  float* h    = dinv + N;
  float* agg  = h + (size_t)N * dHid;
  float* out  = (float*)d_out;

  const long long aggE = (long long)N * dHid;
  const long long outE = (long long)N * dOut;

  // 1) init deg(=1), agg(=0), out(=0)
  {
    long long total = aggE;
    int bs = 256;
    long long gs = (total + bs - 1) / bs;
    gcn_init<<<(unsigned)gs, bs, 0, stream>>>(dinv, agg, out, N, aggE, outE);
  }
  // 2) degree accumulation
  {
    int bs = 256;
    long long gs = (E + bs - 1) / bs;
    gcn_degree<<<(unsigned)gs, bs, 0, stream>>>(dst, dinv, E);
  }
  // 3) dinv = rsqrt(deg)
  gcn_rsqrt<<<(N + 255) / 256, 256, 0, stream>>>(dinv, N);

  // 4) h = x @ W0   (N x dIn) * (dIn x dHid)
  {
    dim3 grid(N / 16, (dHid / 16 + 3) / 4);
    gcn_gemm_wmma<<<grid, 128, 0, stream>>>(x, W0, h, N, dIn, dHid);
  }
  // 5) scatter layer 1: agg[dst] += h[src] * norm
  {
    int bs = 256;  // 8 waves -> 8 edges per block
    long long gs = (E + 7) / 8;
    gcn_scatter<<<(unsigned)gs, bs, 0, stream>>>(h, src, dst, dinv, agg, E, dHid);
  }
  // 6) z = relu(agg + self-loop + b0), in place in agg
  {
    long long total = aggE;
    int bs = 256;
    long long gs = (total + bs - 1) / bs;
    gcn_finalize_relu<<<(unsigned)gs, bs, 0, stream>>>(agg, h, dinv, b0, N, dHid);
  }
  // 7) h2 = z @ W1  (reuse h buffer; dOut <= dHid)
  {
    dim3 grid(N / 16, (dOut / 16 + 3) / 4);
    gcn_gemm_wmma<<<grid, 128, 0, stream>>>(agg, W1, h, N, dHid, dOut);
  }
  // 8) scatter layer 2 into out
  {
    int bs = 256;
    long long gs = (E + 7) / 8;
    gcn_scatter<<<(unsigned)gs, bs, 0, stream>>>(h, src, dst, dinv, out, E, dOut);
  }
  // 9) out += self-loop + b1
  {
    long long total = outE;
    int bs = 256;
    long long gs = (total + bs - 1) / bs;
    gcn_finalize_out<<<(unsigned)gs, bs, 0, stream>>>(out, h, dinv, b1, N, dOut);
  }
}